// GenCONE_85143431676336
// MI455X (gfx1250) — compile-verified
//
#include <hip/hip_runtime.h>
#include <math.h>
#include <stdint.h>

typedef __attribute__((ext_vector_type(2))) float v2f;
typedef __attribute__((ext_vector_type(8))) float v8f;

#define B_   16
#define LE   512
#define LD   128
#define DD   1024

__device__ __forceinline__ v8f wmma4(v2f a, v2f b, v8f c) {
    // V_WMMA_F32_16X16X4_F32 : D = A(16x4) * B(4x16) + C(16x16), fp32 exact
    return __builtin_amdgcn_wmma_f32_16x16x4_f32(
        false, a, false, b, (short)0, c, false, false);
}

// ---------------------------------------------------------------------------
// Kernel 1: s1[b,i] = hc[b,i,:]·w1 + b_sim ;  s2[b,j] = hq[b,j,:]·w2
// One wave per row, shfl reduction (wave32).
// ---------------------------------------------------------------------------
__global__ __launch_bounds__(256) void proj_kernel(
    const float* __restrict__ hc, const float* __restrict__ hq,
    const float* __restrict__ w_sim, const float* __restrict__ b_sim,
    float* __restrict__ s1, float* __restrict__ s2)
{
    int wave = (blockIdx.x * blockDim.x + threadIdx.x) >> 5;
    int lane = threadIdx.x & 31;
    const int nS1 = B_ * LE;
    const int nRows = nS1 + B_ * LD;
    if (wave >= nRows) return;

    const float* row;
    const float* w;
    if (wave < nS1) { row = hc + (size_t)wave * DD;          w = w_sim;      }
    else            { row = hq + (size_t)(wave - nS1) * DD;  w = w_sim + DD; }

    float acc = 0.f;
    #pragma unroll 8
    for (int k = lane; k < DD; k += 32) acc += row[k] * w[k];
    #pragma unroll
    for (int off = 16; off; off >>= 1) acc += __shfl_xor(acc, off, 32);

    if (lane == 0) {
        if (wave < nS1) s1[wave] = acc + b_sim[0];
        else            s2[wave - nS1] = acc;
    }
}

// ---------------------------------------------------------------------------
// Kernel 2: Sim[b,i,j] = sum_d (hc[b,i,d]*w3[d])*hq[b,j,d] + s1[b,i] + s2[b,j]
// Batched GEMM M=512,N=128,K=1024 via V_WMMA_F32_16X16X4_F32.
// ---------------------------------------------------------------------------
__global__ __launch_bounds__(256) void sim_kernel(
    const float* __restrict__ hc, const float* __restrict__ hq,
    const float* __restrict__ w_sim,
    const float* __restrict__ s1, const float* __restrict__ s2,
    float* __restrict__ Sim)
{
    __shared__ float As[16][68];        // (hc * w3) tile, padded stride 68
    __shared__ float Hs[128][68];       // hq tile (rows j, cols k)

    int b  = blockIdx.x >> 5;
    int mt = blockIdx.x & 31;
    int i0 = mt * 16;

    int tid = threadIdx.x;
    int w = tid >> 5, lane = tid & 31;
    int n = lane & 15, g = lane >> 4;
    const float* w3 = w_sim + 2 * DD;

    v8f c = {};

    for (int k0 = 0; k0 < DD; k0 += 64) {
        __syncthreads();
        {   // As: 16x64 floats = 256 float4 -> 1 per thread
            int r  = tid >> 4;
            int cc = (tid & 15) * 4;
            const float* src = hc + ((size_t)(b * LE + i0 + r) * DD + k0 + cc);
            As[r][cc + 0] = src[0] * w3[k0 + cc + 0];
            As[r][cc + 1] = src[1] * w3[k0 + cc + 1];
            As[r][cc + 2] = src[2] * w3[k0 + cc + 2];
            As[r][cc + 3] = src[3] * w3[k0 + cc + 3];
        }
        #pragma unroll
        for (int rep = 0; rep < 8; ++rep) {   // Hs: 128x64 = 2048 float4
            int id = tid + rep * 256;
            int r  = id >> 4;
            int cc = (id & 15) * 4;
            float4 v = *(const float4*)(hq + ((size_t)(b * LD + r) * DD + k0 + cc));
            Hs[r][cc + 0] = v.x; Hs[r][cc + 1] = v.y;
            Hs[r][cc + 2] = v.z; Hs[r][cc + 3] = v.w;
        }
        __syncthreads();

        int j0 = w * 16;
        #pragma unroll
        for (int kk = 0; kk < 64; kk += 4) {
            v2f a, bb;
            a.x  = As[n][kk + 2 * g];
            a.y  = As[n][kk + 2 * g + 1];
            bb.x = Hs[j0 + n][kk + 2 * g];
            bb.y = Hs[j0 + n][kk + 2 * g + 1];
            c = wmma4(a, bb, c);
        }
    }

    int j0 = w * 16;
    #pragma unroll
    for (int r = 0; r < 8; ++r) {
        int i = i0 + r + 8 * g;
        Sim[((size_t)(b * LE) + i) * LD + j0 + n] =
            c[r] + s1[b * LE + i] + s2[b * LD + j0 + n];
    }
}

// ---------------------------------------------------------------------------
// Kernel 3: per-row softmax over j (in place -> A), save pre-softmax row max.
// ---------------------------------------------------------------------------
__global__ __launch_bounds__(256) void softmax_rows_kernel(
    float* __restrict__ Sim, float* __restrict__ rowmax)
{
    int wave = (blockIdx.x * blockDim.x + threadIdx.x) >> 5;
    int lane = threadIdx.x & 31;
    if (wave >= B_ * LE) return;
    float* row = Sim + (size_t)wave * LD;

    float v[4];
    float m = -INFINITY;
    #pragma unroll
    for (int t = 0; t < 4; ++t) { v[t] = row[lane + 32 * t]; m = fmaxf(m, v[t]); }
    #pragma unroll
    for (int off = 16; off; off >>= 1) m = fmaxf(m, __shfl_xor(m, off, 32));

    float s = 0.f;
    #pragma unroll
    for (int t = 0; t < 4; ++t) { v[t] = __expf(v[t] - m); s += v[t]; }
    #pragma unroll
    for (int off = 16; off; off >>= 1) s += __shfl_xor(s, off, 32);

    float inv = 1.0f / s;
    #pragma unroll
    for (int t = 0; t < 4; ++t) row[lane + 32 * t] = v[t] * inv;
    if (lane == 0) rowmax[wave] = m;
}

// ---------------------------------------------------------------------------
// Kernel 4: Batt[b,:] = softmax_i(rowmax[b,:])
// ---------------------------------------------------------------------------
__global__ __launch_bounds__(256) void batt_kernel(
    const float* __restrict__ rowmax, float* __restrict__ Batt)
{
    __shared__ float red[8];
    int b = blockIdx.x;
    int tid = threadIdx.x;
    float v0 = rowmax[b * LE + tid];
    float v1 = rowmax[b * LE + tid + 256];

    float m = fmaxf(v0, v1);
    #pragma unroll
    for (int off = 16; off; off >>= 1) m = fmaxf(m, __shfl_xor(m, off, 32));
    if ((tid & 31) == 0) red[tid >> 5] = m;
    __syncthreads();
    float M = red[0];
    #pragma unroll
    for (int r = 1; r < 8; ++r) M = fmaxf(M, red[r]);

    float e0 = __expf(v0 - M), e1 = __expf(v1 - M);
    float s = e0 + e1;
    #pragma unroll
    for (int off = 16; off; off >>= 1) s += __shfl_xor(s, off, 32);
    __syncthreads();
    if ((tid & 31) == 0) red[tid >> 5] = s;
    __syncthreads();
    float S = 0.f;
    #pragma unroll
    for (int r = 0; r < 8; ++r) S += red[r];
    float inv = 1.0f / S;

    Batt[b * LE + tid]       = e0 * inv;
    Batt[b * LE + tid + 256] = e1 * inv;
}

// ---------------------------------------------------------------------------
// Kernel 5: hc_bar[b,d] = sum_i Batt[b,i] * hc[b,i,d]
// ---------------------------------------------------------------------------
__global__ __launch_bounds__(256) void hcbar_kernel(
    const float* __restrict__ hc, const float* __restrict__ Batt,
    float* __restrict__ hc_bar)
{
    int b = blockIdx.y;
    int d = blockIdx.x * 256 + threadIdx.x;
    const float* base = hc + (size_t)b * LE * DD + d;
    float acc = 0.f;
    #pragma unroll 4
    for (int i = 0; i < LE; ++i) acc += Batt[b * LE + i] * base[(size_t)i * DD];
    hc_bar[b * DD + d] = acc;
}

// ---------------------------------------------------------------------------
// Kernel 6: hq_hat = A(512x128) @ hq(128x1024), per batch, WMMA f32.
// ---------------------------------------------------------------------------
__global__ __launch_bounds__(256) void hqhat_kernel(
    const float* __restrict__ Amat, const float* __restrict__ hq,
    float* __restrict__ hq_hat)
{
    __shared__ float As[16][132];       // full K = 128, padded
    __shared__ float Hs[64][132];       // hq slab: rows k, cols n

    int bidx = blockIdx.x;
    int b    = bidx >> 8;
    int rem  = bidx & 255;
    int mt   = rem >> 3;
    int nb   = rem & 7;
    int i0   = mt * 16;
    int n0   = nb * 128;

    int tid = threadIdx.x;
    int w = tid >> 5, lane = tid & 31;
    int n = lane & 15, g = lane >> 4;

    #pragma unroll
    for (int rep = 0; rep < 2; ++rep) {   // As: 16x128 = 512 float4
        int id = tid + rep * 256;
        int r  = id >> 5;
        int cc = (id & 31) * 4;
        float4 v = *(const float4*)(Amat + ((size_t)(b * LE + i0 + r) * LD + cc));
        As[r][cc + 0] = v.x; As[r][cc + 1] = v.y;
        As[r][cc + 2] = v.z; As[r][cc + 3] = v.w;
    }

    v8f c = {};
    for (int k0 = 0; k0 < LD; k0 += 64) {
        __syncthreads();
        #pragma unroll
        for (int rep = 0; rep < 8; ++rep) {   // Hs: 64x128 = 2048 float4
            int id = tid + rep * 256;
            int r  = id >> 5;
            int cc = (id & 31) * 4;
            float4 v = *(const float4*)(hq + ((size_t)(b * LD + k0 + r) * DD + n0 + cc));
            Hs[r][cc + 0] = v.x; Hs[r][cc + 1] = v.y;
            Hs[r][cc + 2] = v.z; Hs[r][cc + 3] = v.w;
        }
        __syncthreads();

        int nl = w * 16 + n;
        #pragma unroll
        for (int kk = 0; kk < 64; kk += 4) {
            v2f a, bb;
            a.x  = As[n][k0 + kk + 2 * g];
            a.y  = As[n][k0 + kk + 2 * g + 1];
            bb.x = Hs[kk + 2 * g][nl];
            bb.y = Hs[kk + 2 * g + 1][nl];
            c = wmma4(a, bb, c);
        }
    }

    #pragma unroll
    for (int r = 0; r < 8; ++r) {
        int i = i0 + r + 8 * g;
        hq_hat[((size_t)(b * LE) + i) * DD + n0 + w * 16 + n] = c[r];
    }
}

// ---------------------------------------------------------------------------
// Kernel 7: out = feats(8192x4096) @ W_qv(4096x1024) + b_qv.
// feats K-tiles built on the fly (region 0: hc, 1: hq_hat, 2: hc*hq_hat,
// 3: hc*hc_bar). W tile is prefetched with double-buffered CDNA5 async
// loads (GLOBAL_LOAD_ASYNC_TO_LDS_B128, ASYNCcnt) so the bulk of the
// per-stage global traffic overlaps the previous stage's WMMA work.
// ---------------------------------------------------------------------------
__global__ __launch_bounds__(256) void final_kernel(
    const float* __restrict__ hc, const float* __restrict__ hq_hat,
    const float* __restrict__ hc_bar, const float* __restrict__ Wqv,
    const float* __restrict__ b_qv, float* __restrict__ out)
{
    __shared__ float Fs[64][36];        // feats tile 64 x 32 (stride 36)
    __shared__ float Ws[2][32][136];    // double-buffered W tile 32 x 128

    int row0 = blockIdx.x * 64;
    int b    = row0 / LE;
    int i0   = row0 % LE;
    int n0   = blockIdx.y * 128;

    int tid = threadIdx.x;
    int w = tid >> 5, lane = tid & 31;
    int n = lane & 15, g = lane >> 4;
    int msub = w >> 1, nhalf = w & 1;

    // Per-thread async-copy coordinates for the W tile (4 x b128 per thread).
    int wr[4], wc[4];
    #pragma unroll
    for (int rep = 0; rep < 4; ++rep) {
        int id = tid + rep * 256;
        wr[rep] = id >> 5;            // k-row within tile, 0..31
        wc[rep] = (id & 31) * 4;      // col within 128-wide window
    }

    // Issue one stage's W tile: memory -> LDS, no VGPR staging (ASYNCcnt).
    auto issueW = [&](int stage) {
        int buf = stage & 1;
        #pragma unroll
        for (int rep = 0; rep < 4; ++rep) {
            uint64_t gaddr = (uint64_t)(uintptr_t)(
                Wqv + ((size_t)(stage * 32 + wr[rep])) * DD + n0 + wc[rep]);
            uint32_t laddr = (uint32_t)(uintptr_t)&Ws[buf][wr[rep]][wc[rep]];
            asm volatile("global_load_async_to_lds_b128 %0, %1, off"
                         :: "v"(laddr), "v"(gaddr) : "memory");
        }
    };

    v8f c[4] = {};

    issueW(0);                               // prologue prefetch

    for (int s = 0; s < 128; ++s) {
        int k0     = s * 32;
        int region = k0 >> 10;
        int koff   = k0 & (DD - 1);

        __syncthreads();                     // Fs + Ws[(s+1)&1] free now
        if (s + 1 < 128) issueW(s + 1);      // prefetch next W tile

        #pragma unroll
        for (int rep = 0; rep < 2; ++rep) {  // feats: 64x32 = 512 float4
            int id = tid + rep * 256;
            int r  = id >> 3;
            int cc = (id & 7) * 4;
            size_t base = ((size_t)(b * LE + i0 + r)) * DD + koff + cc;
            float4 v;
            if (region == 0) {
                v = *(const float4*)(hc + base);
            } else if (region == 1) {
                v = *(const float4*)(hq_hat + base);
            } else if (region == 2) {
                float4 x = *(const float4*)(hc + base);
                float4 y = *(const float4*)(hq_hat + base);
                v = make_float4(x.x * y.x, x.y * y.y, x.z * y.z, x.w * y.w);
            } else {
                float4 x = *(const float4*)(hc + base);
                float4 y = *(const float4*)(hc_bar + (size_t)b * DD + koff + cc);
                v = make_float4(x.x * y.x, x.y * y.y, x.z * y.z, x.w * y.w);
            }
            Fs[r][cc + 0] = v.x; Fs[r][cc + 1] = v.y;
            Fs[r][cc + 2] = v.z; Fs[r][cc + 3] = v.w;
        }

        // Wait for *this* stage's W tile (next stage's 4 may stay in flight).
        if (s + 1 < 128) asm volatile("s_wait_asynccnt 0x4" ::: "memory");
        else             asm volatile("s_wait_asynccnt 0x0" ::: "memory");
        __syncthreads();

        #pragma unroll
        for (int kk = 0; kk < 32; kk += 4) {
            v2f a;
            a.x = Fs[16 * msub + n][kk + 2 * g];
            a.y = Fs[16 * msub + n][kk + 2 * g + 1];
            #pragma unroll
            for (int t = 0; t < 4; ++t) {
                v2f bb;
                int nl = 64 * nhalf + 16 * t + n;
                bb.x = Ws[s & 1][kk + 2 * g][nl];
                bb.y = Ws[s & 1][kk + 2 * g + 1][nl];
                c[t] = wmma4(a, bb, c[t]);
            }
        }
    }

    #pragma unroll
    for (int t = 0; t < 4; ++t) {
        int ncol = n0 + 64 * nhalf + 16 * t + n;
        float bq = b_qv[ncol];
        #pragma unroll
        for (int r = 0; r < 8; ++r) {
            int i = i0 + 16 * msub + r + 8 * g;
            out[((size_t)(b * LE) + i) * DD + ncol] = c[t][r] + bq;
        }
    }
}

// ---------------------------------------------------------------------------
extern "C" void kernel_launch(void* const* d_in, const int* in_sizes, int n_in,
                              void* d_out, int out_size, void* d_ws, size_t ws_size,
                              hipStream_t stream) {
    const float* hc    = (const float*)d_in[0];   // (16,512,1024)
    const float* hq    = (const float*)d_in[1];   // (16,128,1024)
    const float* w_sim = (const float*)d_in[2];   // (3072,)
    const float* b_sim = (const float*)d_in[3];   // (1,)
    const float* W_qv  = (const float*)d_in[4];   // (4096,1024)
    const float* b_qv  = (const float*)d_in[5];   // (1024,)
    float* out = (float*)d_out;

    float* ws = (float*)d_ws;
    float* hq_hat = ws;                              // 8,388,608
    float* Sim    = hq_hat + (size_t)B_ * LE * DD;   // 1,048,576 (becomes A)
    float* s1     = Sim    + (size_t)B_ * LE * LD;   // 8192
    float* s2     = s1     + B_ * LE;                // 2048
    float* rowmax = s2     + B_ * LD;                // 8192
    float* Batt   = rowmax + B_ * LE;                // 8192
    float* hc_bar = Batt   + B_ * LE;                // 16384
    // total workspace: ~37.9 MB

    proj_kernel<<<1280, 256, 0, stream>>>(hc, hq, w_sim, b_sim, s1, s2);
    sim_kernel<<<B_ * 32, 256, 0, stream>>>(hc, hq, w_sim, s1, s2, Sim);
    softmax_rows_kernel<<<1024, 256, 0, stream>>>(Sim, rowmax);
    batt_kernel<<<B_, 256, 0, stream>>>(rowmax, Batt);
    hcbar_kernel<<<dim3(DD / 256, B_), 256, 0, stream>>>(hc, Batt, hc_bar);
    hqhat_kernel<<<4096, 256, 0, stream>>>(Sim, hq, hq_hat);
    final_kernel<<<dim3(128, 8), 256, 0, stream>>>(hc, hq_hat, hc_bar, W_qv, b_qv, out);
}